// ICRcriterion_61297773248742
// MI455X (gfx1250) — compile-verified
//
#include <hip/hip_runtime.h>
#include <hip/hip_bf16.h>
#include <cstdint>

// ICR criterion: softmax CE with cluster-neighbour probability pooling.
// One workgroup per batch row. The 128KB fp32 logit row is DMA'd into LDS by
// the Tensor Data Mover as TWO 64KB descriptors so the row-max pass on the
// first half overlaps the DMA of the second half (TENSORcnt pipelining).
// Max pass, exp-sum pass, and the ragged 64-wide neighbour gather all run
// out of LDS, so HBM sees x exactly once: 512MB @ 23.3TB/s ~= 22us floor.
// TDM loads use th:TH_LOAD_NT so the one-shot stream doesn't rinse the 192MB
// L2 that holds the reused y/position/neighbours tables.

typedef __attribute__((ext_vector_type(4))) unsigned int u32x4;
typedef __attribute__((ext_vector_type(8))) unsigned int u32x8;

#define C_DIM 32768   // classes per row (128 KB fp32)
#define K_DIM 64      // max neighbours
#define BLK   256     // 8 waves (wave32)
#define NWAVE (BLK / 32)
#define HALF  (C_DIM / 2)      // elements per TDM descriptor
#define HALF4 (HALF / 4)       // float4 count per half

__device__ __forceinline__ void tdm_load_row_half(const float* gsrc,
                                                  const float* lds_dst,
                                                  int nelem)
{
    const uint64_t gaddr   = (uint64_t)(uintptr_t)gsrc;
    const uint32_t lds_off = (uint32_t)(uintptr_t)lds_dst; // flat->LDS byte offset

    // D# group 0: count=1 | lds_addr | global_addr[56:0] | type=2
    u32x4 g0;
    g0[0] = 1u;
    g0[1] = lds_off;
    g0[2] = (uint32_t)(gaddr & 0xFFFFFFFFull);
    g0[3] = (uint32_t)((gaddr >> 32) & 0x01FFFFFFull) | (2u << 30);

    // D# group 1: 1-D tile of nelem fp32 elements
    u32x8 g1;
    g1[0] = (2u << 16);                                    // data_size = 4 B
    g1[1] = ((uint32_t)(nelem & 0xFFFF)) << 16;            // tensor_dim0[15:0]
    g1[2] = ((uint32_t)(nelem >> 16) & 0xFFFFu) | (1u << 16); // dim0[31:16], dim1=1
    g1[3] = ((uint32_t)(nelem & 0xFFFF)) << 16;            // tile_dim0
    g1[4] = 0u;                                            // tile_dim1/2 unused
    g1[5] = (uint32_t)nelem;                               // tensor_dim0_stride
    g1[6] = 0u;
    g1[7] = 0u;

    asm volatile("tensor_load_to_lds %0, %1 th:TH_LOAD_NT"
                 :: "s"(g0), "s"(g1)
                 : "memory");
}

__device__ __forceinline__ float wave_max(float v) {
#pragma unroll
    for (int off = 16; off > 0; off >>= 1)
        v = fmaxf(v, __shfl_xor(v, off, 32));
    return v;
}
__device__ __forceinline__ float wave_sum(float v) {
#pragma unroll
    for (int off = 16; off > 0; off >>= 1)
        v += __shfl_xor(v, off, 32);
    return v;
}

__global__ __launch_bounds__(BLK) void icr_row_kernel(
    const float* __restrict__ x,
    const int* __restrict__ y,
    const int* __restrict__ position,
    const int* __restrict__ neighbours,
    const unsigned char* __restrict__ nmask,   // bool array, 1 byte/elem
    float* __restrict__ partial)
{
    __shared__ float lds_row[C_DIM];   // 128 KB row buffer (TDM destination)
    __shared__ float red[NWAVE];

    const int b    = blockIdx.x;
    const int tid  = threadIdx.x;
    const int wid  = tid >> 5;
    const int lane = tid & 31;

    // ---- TDM: issue both half-row DMAs (wave 0 only; TDM ignores EXEC) ----
    if (tid < 32) {
        tdm_load_row_half(x + (size_t)b * C_DIM,        &lds_row[0],    HALF);
        tdm_load_row_half(x + (size_t)b * C_DIM + HALF, &lds_row[HALF], HALF);
        __builtin_amdgcn_s_wait_tensorcnt(1);   // first 64KB landed
    }
    __syncthreads();

    const float4* row4 = (const float4*)lds_row;

    // ---- Pass 1a: max over half 0 (overlaps half-1 DMA) ----
    float m = -3.402823466e+38f;
#pragma unroll 4
    for (int i = tid; i < HALF4; i += BLK) {
        float4 v = row4[i];
        m = fmaxf(m, fmaxf(fmaxf(v.x, v.y), fmaxf(v.z, v.w)));
    }

    if (tid < 32) __builtin_amdgcn_s_wait_tensorcnt(0);   // second 64KB landed
    __syncthreads();

    // ---- Pass 1b: max over half 1 ----
#pragma unroll 4
    for (int i = HALF4 + tid; i < 2 * HALF4; i += BLK) {
        float4 v = row4[i];
        m = fmaxf(m, fmaxf(fmaxf(v.x, v.y), fmaxf(v.z, v.w)));
    }
    m = wave_max(m);
    if (lane == 0) red[wid] = m;
    __syncthreads();
    float rowmax = red[0];
#pragma unroll
    for (int i = 1; i < NWAVE; ++i) rowmax = fmaxf(rowmax, red[i]);
    __syncthreads();

    // ---- Pass 2: sum of exp over full row (from LDS) ----
    float s = 0.f;
#pragma unroll 4
    for (int i = tid; i < C_DIM / 4; i += BLK) {
        float4 v = row4[i];
        s += __expf(v.x - rowmax) + __expf(v.y - rowmax)
           + __expf(v.z - rowmax) + __expf(v.w - rowmax);
    }
    s = wave_sum(s);
    if (lane == 0) red[wid] = s;
    __syncthreads();
    float sumexp = red[0];
#pragma unroll
    for (int i = 1; i < NWAVE; ++i) sumexp += red[i];
    __syncthreads();

    // ---- Ragged neighbour gather (random access served by LDS) ----
    const int  yb      = y[b];
    const int  pos     = position[yb];
    const bool cluster = (pos >= 0);
    const int  list    = cluster ? pos : 0;

    float nsum = 0.f;
    if (cluster && tid < K_DIM) {
        const int nb           = neighbours[(size_t)list * K_DIM + tid];
        const unsigned char mk = nmask[(size_t)list * K_DIM + tid];
        if (mk) nsum = __expf(lds_row[nb] - rowmax);
    }
    nsum = wave_sum(nsum);
    if (lane == 0) red[wid] = nsum;
    __syncthreads();

    if (tid == 0) {
        float nbr = 0.f;
#pragma unroll
        for (int i = 0; i < NWAVE; ++i) nbr += red[i];
        const float e_label = __expf(lds_row[yb] - rowmax);
        // -log((e_label + nbr_sum)/sumexp)
        partial[b] = __logf(sumexp) - __logf(e_label + nbr);
    }
}

// Deterministic fixed-order final reduction (no float atomics).
__global__ __launch_bounds__(BLK) void icr_final_reduce(
    const float* __restrict__ partial, float* __restrict__ out, int n)
{
    __shared__ float red[NWAVE];
    float s = 0.f;
    for (int i = threadIdx.x; i < n; i += BLK) s += partial[i];
    s = wave_sum(s);
    if ((threadIdx.x & 31) == 0) red[threadIdx.x >> 5] = s;
    __syncthreads();
    if (threadIdx.x == 0) {
        float t = 0.f;
#pragma unroll
        for (int i = 0; i < NWAVE; ++i) t += red[i];
        out[0] = t / (float)n;
    }
}

extern "C" void kernel_launch(void* const* d_in, const int* in_sizes, int n_in,
                              void* d_out, int out_size, void* d_ws, size_t ws_size,
                              hipStream_t stream)
{
    const float*         x          = (const float*)d_in[0];
    const int*           y          = (const int*)d_in[1];
    const int*           position   = (const int*)d_in[2];
    const int*           neighbours = (const int*)d_in[3];
    const unsigned char* nmask      = (const unsigned char*)d_in[4];
    float*               out        = (float*)d_out;

    const int B = in_sizes[1];                 // 4096
    float* partial = (float*)d_ws;             // B floats of scratch

    icr_row_kernel<<<B, BLK, 0, stream>>>(x, y, position, neighbours, nmask, partial);
    icr_final_reduce<<<1, BLK, 0, stream>>>(partial, out, B);
}